// GraphAttentionLayer_867583394401
// MI455X (gfx1250) — compile-verified
//
#include <hip/hip_runtime.h>
#include <hip/hip_bf16.h>

typedef __attribute__((ext_vector_type(2))) float v2f;
typedef __attribute__((ext_vector_type(8))) float v8f;
typedef __attribute__((ext_vector_type(4))) unsigned int u32x4;
typedef __attribute__((ext_vector_type(8))) int i32x8;
typedef __attribute__((ext_vector_type(4))) int i32x4;

#define B_DIM   8
#define N_DIM   2048
#define F_IN    512
#define F_OUT   256
#define ALPHA   0.2f
#define NEG_INF (-9.0e15f)

// LDS row stride for the staged Wh K-slab: 256 + 4 pad DWORDs (bank-conflict free)
#define BSTRIDE 260

// ---------------------------------------------------------------------------
// TDM descriptor: load a 16(row) x 256(col) f32 tile of Wh into LDS with
// 4-DWORD padding after each 256-DWORD row (ISA cdna5 §8: D# groups).
// 6-arg builtin form (clang-23 / therock-10.0 headers).
// ---------------------------------------------------------------------------
__device__ __forceinline__ void tdm_load_tile(unsigned lds_off,
                                              unsigned long long gaddr) {
    u32x4 g0;
    g0.x = 1u;                                          // count=1, user D#
    g0.y = lds_off;                                     // lds_addr [63:32]
    g0.z = (unsigned)(gaddr & 0xFFFFFFFFull);           // global_addr[31:0]
    g0.w = (unsigned)((gaddr >> 32) & 0x01FFFFFFull)    // global_addr[56:32]
           | 0x80000000u;                               // type=2 at [127:126]
    i32x8 g1;
    g1[0] = (int)((2u << 16)    // data_size = 4B
                | (1u << 20)    // pad_enable
                | (7u << 22)    // pad_interval: 256 DWORDs
                | (3u << 25));  // pad_amount: 4 DWORDs
    g1[1] = (int)(256u << 16);  // tensor_dim0 = 256 (bits [79:48])
    g1[2] = (int)(2048u << 16); // tensor_dim1 = 2048 (bits [111:80])
    g1[3] = (int)(256u << 16);  // tile_dim0 = 256 (bits [127:112])
    g1[4] = 16;                 // tile_dim1 = 16
    g1[5] = 256;                // tensor_dim0_stride = 256
    g1[6] = 0;
    g1[7] = 0;
    i32x4 z4 = {0, 0, 0, 0};
    i32x8 z8 = {0, 0, 0, 0, 0, 0, 0, 0};
    __builtin_amdgcn_tensor_load_to_lds(g0, g1, z4, z4, z8, 0);
}

// ---------------------------------------------------------------------------
// Kernel 1: Wh = h @ W  (WMMA f32 16x16x4), fused s1 = Wh@a1, s2 = Wh@a2.
// ---------------------------------------------------------------------------
__global__ __launch_bounds__(256) void k_wh(const float* __restrict__ h,
                                            const float* __restrict__ W,
                                            const float* __restrict__ a,
                                            float* __restrict__ Wh,
                                            float* __restrict__ s1g,
                                            float* __restrict__ s2g) {
    __shared__ float hA[16 * 516];
    __shared__ float sAcc[2][16];

    const int t    = threadIdx.x;
    const int row0 = blockIdx.x * 16;

    if (t < 32) sAcc[t >> 4][t & 15] = 0.0f;

    for (int idx = t; idx < 16 * 128; idx += 256) {
        const int r  = idx >> 7;
        const int kq = (idx & 127) << 2;
        const float4 v = *(const float4*)(h + (size_t)(row0 + r) * F_IN + kq);
        float* dst = &hA[r * 516 + kq];
        dst[0] = v.x; dst[1] = v.y; dst[2] = v.z; dst[3] = v.w;
    }
    __syncthreads();

    const int wave = t >> 5;
    const int lane = t & 31;
    const int m    = lane & 15;
    const int hh   = lane >> 4;
    const int col0 = wave * 32;

    v8f c0 = {};
    v8f c1 = {};

    for (int k = 0; k < F_IN; k += 4) {
        const int kk = k + 2 * hh;
        v2f aA;
        aA.x = hA[m * 516 + kk];
        aA.y = hA[m * 516 + kk + 1];

        const float* Wp = W + (size_t)kk * F_OUT;
        v2f b0, b1;
        b0.x = Wp[col0 + m];      b0.y = Wp[F_OUT + col0 + m];
        b1.x = Wp[col0 + 16 + m]; b1.y = Wp[F_OUT + col0 + 16 + m];

        c0 = __builtin_amdgcn_wmma_f32_16x16x4_f32(false, aA, false, b0,
                                                   (short)0, c0, false, false);
        c1 = __builtin_amdgcn_wmma_f32_16x16x4_f32(false, aA, false, b1,
                                                   (short)0, c1, false, false);
    }

    const float a1c0 = a[col0 + m],         a1c1 = a[col0 + 16 + m];
    const float a2c0 = a[F_OUT + col0 + m], a2c1 = a[F_OUT + col0 + 16 + m];
#pragma unroll
    for (int r = 0; r < 8; ++r) {
        const int rr = r + 8 * hh;
        const float w0 = c0[r];
        const float w1 = c1[r];
        Wh[(size_t)(row0 + rr) * F_OUT + col0 + m]      = w0;
        Wh[(size_t)(row0 + rr) * F_OUT + col0 + 16 + m] = w1;
        atomicAdd(&sAcc[0][rr], w0 * a1c0 + w1 * a1c1);
        atomicAdd(&sAcc[1][rr], w0 * a2c0 + w1 * a2c1);
    }
    __syncthreads();
    if (t < 16) {
        s1g[row0 + t] = sAcc[0][t];
        s2g[row0 + t] = sAcc[1][t];
    }
}

// ---------------------------------------------------------------------------
// Kernel 2: per-row online softmax stats over masked leaky(s1_i + s2_j).
// ---------------------------------------------------------------------------
__global__ __launch_bounds__(256) void k_stats(const float* __restrict__ s1,
                                               const float* __restrict__ s2,
                                               const int* __restrict__ adj,
                                               float* __restrict__ rmax,
                                               float* __restrict__ rrcp) {
    const int wid  = blockIdx.x * 8 + (threadIdx.x >> 5);
    const int lane = threadIdx.x & 31;
    const int b    = wid >> 11;
    const int i    = wid & (N_DIM - 1);

    const float si = s1[wid];
    const int*   arow = adj + ((size_t)b * N_DIM + i) * N_DIM;
    const float* s2b  = s2 + b * N_DIM;

    float mx = -3.0e38f;
    float l  = 0.0f;
    for (int j = lane; j < N_DIM; j += 32) {
        float e = si + s2b[j];
        e = e > 0.0f ? e : ALPHA * e;
        const float v = (arow[j] > 0) ? e : NEG_INF;
        if (v > mx) { l = l * __expf(mx - v) + 1.0f; mx = v; }
        else        { l += __expf(v - mx); }
    }
    for (int off = 16; off; off >>= 1) {
        const float m2 = __shfl_xor(mx, off, 32);
        const float l2 = __shfl_xor(l,  off, 32);
        const float M  = fmaxf(mx, m2);
        l  = l * __expf(mx - M) + l2 * __expf(m2 - M);
        mx = M;
    }
    if (lane == 0) { rmax[wid] = mx; rrcp[wid] = 1.0f / l; }
}

// ---------------------------------------------------------------------------
// Kernel 3: h' = softmax(mask(leaky(rank1))) @ Wh, fused ELU.
// Block = 128 rows x 256 cols, 1024 threads (32 waves: 8 row-groups x 4
// col-groups). Wh K-slabs (16x256) staged to LDS via double-buffered TDM
// (tensor_load_to_lds + s_wait_tensorcnt); P tile (128x16) built per K-step.
// ---------------------------------------------------------------------------
__global__ __launch_bounds__(1024) void k_out(const float* __restrict__ Wh,
                                              const int* __restrict__ adj,
                                              const float* __restrict__ s1,
                                              const float* __restrict__ s2,
                                              const float* __restrict__ rmax,
                                              const float* __restrict__ rrcp,
                                              float* __restrict__ out) {
    __shared__ float bufB[2][16 * BSTRIDE];   // staged Wh K-slabs (padded rows)
    __shared__ float P[128 * 18];             // probability tile

    const int b  = blockIdx.y;
    const int i0 = blockIdx.x * 128;
    const int t  = threadIdx.x;

    // P-building role: rows (ti, ti+64), col tj
    const int ti = t >> 4;
    const int tj = t & 15;
    // WMMA role
    const int wave = t >> 5;           // 0..31
    const int rg   = wave >> 2;        // row group 0..7 (16 rows each)
    const int cg   = wave & 3;         // col group 0..3 (64 cols each)
    const int lane = t & 31;
    const int m    = lane & 15;
    const int hh   = lane >> 4;
    const int col0 = cg * 64;

    const size_t rowbase = (size_t)b * N_DIM + i0;
    const float smi0 = s1[rowbase + ti],        smi1 = s1[rowbase + ti + 64];
    const float rm0  = rmax[rowbase + ti],      rm1  = rmax[rowbase + ti + 64];
    const float rc0  = rrcp[rowbase + ti],      rc1  = rrcp[rowbase + ti + 64];
    const int* arow0 = adj + (rowbase + ti) * N_DIM;
    const int* arow1 = adj + (rowbase + ti + 64) * N_DIM;
    const float* s2b = s2 + b * N_DIM;
    const float* WhB = Wh + (size_t)b * N_DIM * F_OUT;

    const unsigned buf_off[2] = {
        (unsigned)(uintptr_t)&bufB[0][0],
        (unsigned)(uintptr_t)&bufB[1][0]
    };
    const unsigned long long gbase = (unsigned long long)(uintptr_t)WhB;

    v8f acc[4] = {v8f{}, v8f{}, v8f{}, v8f{}};

    // prologue: stage K-slab 0
    if (t < 32) tdm_load_tile(buf_off[0], gbase);

    for (int j0 = 0; j0 < N_DIM; j0 += 16) {
        const int cur = (j0 >> 4) & 1;
        // issue next slab, then wait for the current one (TDM is in-order)
        if (t < 32) {
            if (j0 + 16 < N_DIM) {
                tdm_load_tile(buf_off[cur ^ 1],
                              gbase + (unsigned long long)(j0 + 16) * F_OUT * 4);
                __builtin_amdgcn_s_wait_tensorcnt(1);
            } else {
                __builtin_amdgcn_s_wait_tensorcnt(0);
            }
        }

        if (j0 + 16 < N_DIM)
            __builtin_prefetch(arow0 + j0 + 16, 0, 0);

        const float sj = s2b[j0 + tj];
        float e0 = smi0 + sj;  e0 = e0 > 0.0f ? e0 : ALPHA * e0;
        float e1 = smi1 + sj;  e1 = e1 > 0.0f ? e1 : ALPHA * e1;
        const float v0 = (arow0[j0 + tj] > 0) ? e0 : NEG_INF;
        const float v1 = (arow1[j0 + tj] > 0) ? e1 : NEG_INF;
        const float p0 = __expf(v0 - rm0) * rc0;
        const float p1 = __expf(v1 - rm1) * rc1;

        __syncthreads();                     // TDM slab visible; prev P consumed
        P[ti * 18 + tj]        = p0;
        P[(ti + 64) * 18 + tj] = p1;
        __syncthreads();                     // P visible

        const float* Bbuf = &bufB[cur][0];
#pragma unroll
        for (int ks = 0; ks < 16; ks += 4) {
            const int kk = ks + 2 * hh;
            v2f aA;
            aA.x = P[(rg * 16 + m) * 18 + kk];
            aA.y = P[(rg * 16 + m) * 18 + kk + 1];
#pragma unroll
            for (int ct = 0; ct < 4; ++ct) {
                v2f bb;
                bb.x = Bbuf[kk * BSTRIDE + col0 + ct * 16 + m];
                bb.y = Bbuf[(kk + 1) * BSTRIDE + col0 + ct * 16 + m];
                acc[ct] = __builtin_amdgcn_wmma_f32_16x16x4_f32(
                    false, aA, false, bb, (short)0, acc[ct], false, false);
            }
        }
        __syncthreads();                     // done with bufB[cur] & P
    }

    // epilogue: ELU + store
#pragma unroll
    for (int ct = 0; ct < 4; ++ct) {
#pragma unroll
        for (int r = 0; r < 8; ++r) {
            const int row = i0 + rg * 16 + r + 8 * hh;
            float x = acc[ct][r];
            x = x > 0.0f ? x : (__expf(x) - 1.0f);
            out[((size_t)b * N_DIM + row) * F_OUT + col0 + ct * 16 + m] = x;
        }
    }
}

// ---------------------------------------------------------------------------
extern "C" void kernel_launch(void* const* d_in, const int* in_sizes, int n_in,
                              void* d_out, int out_size, void* d_ws, size_t ws_size,
                              hipStream_t stream) {
    const float* h   = (const float*)d_in[0];
    const int*   adj = (const int*)d_in[1];
    const float* W   = (const float*)d_in[2];
    const float* a   = (const float*)d_in[3];
    float* out = (float*)d_out;

    float* Wh   = (float*)d_ws;
    float* s1   = Wh + (size_t)B_DIM * N_DIM * F_OUT;
    float* s2   = s1 + B_DIM * N_DIM;
    float* rmax = s2 + B_DIM * N_DIM;
    float* rrcp = rmax + B_DIM * N_DIM;

    const int rows = B_DIM * N_DIM;

    k_wh<<<dim3(rows / 16), dim3(256), 0, stream>>>(h, W, a, Wh, s1, s2);
    k_stats<<<dim3(rows / 8), dim3(256), 0, stream>>>(s1, s2, adj, rmax, rrcp);
    k_out<<<dim3(N_DIM / 128, B_DIM), dim3(1024), 0, stream>>>(Wh, adj, s1, s2,
                                                               rmax, rrcp, out);
}